// EpisodicMemory_43112881717513
// MI455X (gfx1250) — compile-verified
//
#include <hip/hip_runtime.h>
#include <math.h>

typedef __attribute__((ext_vector_type(2))) float v2f;
typedef __attribute__((ext_vector_type(8))) float v8f;

#define BS      32
#define M_SLOTS 32768
#define D_IN    1024
#define D_EMB   128
#define K_RET   32
#define CHUNK   2048
#define NBLK_M  (M_SLOTS / CHUNK)   // 16

static __device__ inline v8f wmma_step(v2f a, v2f b, v8f acc) {
  // V_WMMA_F32_16X16X4_F32 : D = A(16x4) * B(4x16) + C, fp32 throughout
  return __builtin_amdgcn_wmma_f32_16x16x4_f32(false, a, false, b, (short)0, acc,
                                               false, false);
}

// -------------------------------------------------------------------------
// Kernel 1: q = unit_normalize(concat(x,y) @ Wq_em + bq)   [32,128]
//           q_cross = x @ Wq_cross + bc                    [32,128]
// One block, 16 waves; wave -> one 16x16 output tile (2 M-tiles x 8 N-tiles).
// -------------------------------------------------------------------------
__global__ __launch_bounds__(512)
void proj_kernel(const float* __restrict__ x, const float* __restrict__ y,
                 const float* __restrict__ Wq, const float* __restrict__ bq,
                 const float* __restrict__ Wc, const float* __restrict__ bc,
                 float* __restrict__ q_out, float* __restrict__ qc_out) {
  __shared__ __align__(16) float qtmp[BS][D_EMB];
  __shared__ float rn[BS];
  const int tid  = threadIdx.x;
  const int wave = tid >> 5;
  const int lane = tid & 31;
  const int half = lane >> 4;
  const int l16  = lane & 15;
  const int mt   = wave & 1;
  const int nt   = wave >> 1;
  const int m    = mt * 16 + l16;
  const int n    = nt * 16 + l16;

  // ---- q projection: K = 2048 (x then y_wm) -------------------------------
  {
    v8f acc = {};
    for (int k = 0; k < 2 * D_IN; k += 4) {
      const float* arow = (k < D_IN) ? (x + (size_t)m * D_IN + k)
                                     : (y + (size_t)m * D_IN + (k - D_IN));
      v2f a, b;
      a.x = arow[2 * half + 0];
      a.y = arow[2 * half + 1];
      b.x = Wq[(size_t)(k + 2 * half + 0) * D_EMB + n];
      b.y = Wq[(size_t)(k + 2 * half + 1) * D_EMB + n];
      acc = wmma_step(a, b, acc);
    }
    const float bias = bq[n];
#pragma unroll
    for (int r = 0; r < 8; ++r)
      qtmp[mt * 16 + half * 8 + r][n] = acc[r] + bias;
  }
  __syncthreads();
  if (tid < BS) {
    float s = 0.f;
    for (int d = 0; d < D_EMB; ++d) { float v = qtmp[tid][d]; s += v * v; }
    rn[tid] = rsqrtf(s + 1e-12f);
  }
  __syncthreads();
  for (int i = tid; i < BS * D_EMB; i += 512)
    q_out[i] = qtmp[i / D_EMB][i % D_EMB] * rn[i / D_EMB];

  // ---- q_cross projection: K = 1024, raw ----------------------------------
  {
    v8f acc = {};
    for (int k = 0; k < D_IN; k += 4) {
      v2f a, b;
      a.x = x[(size_t)m * D_IN + k + 2 * half + 0];
      a.y = x[(size_t)m * D_IN + k + 2 * half + 1];
      b.x = Wc[(size_t)(k + 2 * half + 0) * D_EMB + n];
      b.y = Wc[(size_t)(k + 2 * half + 1) * D_EMB + n];
      acc = wmma_step(a, b, acc);
    }
    const float bias = bc[n];
#pragma unroll
    for (int r = 0; r < 8; ++r)
      qc_out[(size_t)(mt * 16 + half * 8 + r) * D_EMB + n] = acc[r] + bias;
  }
}

// -------------------------------------------------------------------------
// Kernel 2: streaming scores + per-block top-32.
// Grid (NBLK_M, BS), 256 threads. Each wave: lane l holds q[4l..4l+3]; one
// float4 load per lane covers a full 512B em_K slot row (coalesced).
// -------------------------------------------------------------------------
__global__ __launch_bounds__(256)
void scores_topk_kernel(const float* __restrict__ q,
                        const float* __restrict__ em_K,
                        const float* __restrict__ em_S,
                        float* __restrict__ cand_s, int* __restrict__ cand_i) {
  __shared__ __align__(16) float sc[CHUNK];
  __shared__ __align__(16) float qs[D_EMB];
  __shared__ float red_v[256];
  __shared__ int   red_i[256];
  const int b    = blockIdx.y;
  const int mb   = blockIdx.x;
  const int m0   = mb * CHUNK;
  const int tid  = threadIdx.x;
  const int wave = tid >> 5;
  const int lane = tid & 31;

  if (tid < D_EMB) qs[tid] = q[b * D_EMB + tid];
  __syncthreads();
  const float4 q4 = *reinterpret_cast<const float4*>(&qs[lane * 4]);
  const size_t kbase = (size_t)b * M_SLOTS * D_EMB;

  for (int it = 0; it < CHUNK / 32; ++it) {       // 64 iterations
    const int sbase = m0 + (it * 8 + wave) * 4;   // 4 consecutive slots / wave
    float sum0, sum1, sum2, sum3;
#pragma unroll
    for (int j = 0; j < 4; ++j) {
      const float4 k4 = *reinterpret_cast<const float4*>(
          em_K + kbase + (size_t)(sbase + j) * D_EMB + lane * 4);
      // prefetch the row this wave will touch 8 iterations from now
      __builtin_prefetch(em_K + kbase + (size_t)(sbase + 256 + j) * D_EMB + lane * 4, 0, 0);
      float p = q4.x * k4.x + q4.y * k4.y + q4.z * k4.z + q4.w * k4.w;
#pragma unroll
      for (int msk = 16; msk >= 1; msk >>= 1) p += __shfl_xor(p, msk, 32);
      if (j == 0) sum0 = p; else if (j == 1) sum1 = p;
      else if (j == 2) sum2 = p; else sum3 = p;
    }
    if (lane < 4) {
      const int s = sbase + lane;
      const float v = (lane == 0) ? sum0 : (lane == 1) ? sum1
                    : (lane == 2) ? sum2 : sum3;
      const float act = em_S[(size_t)b * M_SLOTS + s];
      sc[s - m0] = (act > 0.f) ? v : -INFINITY;
    }
  }
  __syncthreads();

  // iterative argmax: block-local top-32 of 2048
  float* outs = cand_s + ((size_t)b * NBLK_M + mb) * K_RET;
  int*   outi = cand_i + ((size_t)b * NBLK_M + mb) * K_RET;
  for (int sel = 0; sel < K_RET; ++sel) {
    float bv = -INFINITY; int bi = -1;
    for (int j = tid; j < CHUNK; j += 256) {
      float v = sc[j];
      if (v > bv) { bv = v; bi = j; }
    }
    red_v[tid] = bv; red_i[tid] = bi;
    __syncthreads();
    for (int s = 128; s > 0; s >>= 1) {
      if (tid < s && red_v[tid + s] > red_v[tid]) {
        red_v[tid] = red_v[tid + s]; red_i[tid] = red_i[tid + s];
      }
      __syncthreads();
    }
    if (tid == 0) {
      int w = red_i[0];
      outs[sel] = red_v[0];
      outi[sel] = (w >= 0) ? (m0 + w) : -1;
      if (w >= 0) sc[w] = -INFINITY;
    }
    __syncthreads();
  }
}

// -------------------------------------------------------------------------
// Kernel 3: merge per-block candidates -> global top-32 per batch row.
// -------------------------------------------------------------------------
__global__ __launch_bounds__(256)
void merge_topk_kernel(const float* __restrict__ cand_s, const int* __restrict__ cand_i,
                       float* __restrict__ top_s, int* __restrict__ top_i) {
  const int N = NBLK_M * K_RET;   // 512
  __shared__ float sc[NBLK_M * K_RET];
  __shared__ int   si[NBLK_M * K_RET];
  __shared__ float red_v[256];
  __shared__ int   red_i[256];
  const int b = blockIdx.x;
  const int tid = threadIdx.x;
  for (int j = tid; j < N; j += 256) {
    sc[j] = cand_s[(size_t)b * N + j];
    si[j] = cand_i[(size_t)b * N + j];
  }
  __syncthreads();
  for (int sel = 0; sel < K_RET; ++sel) {
    float bv = -INFINITY; int bi = -1;
    for (int j = tid; j < N; j += 256) {
      float v = sc[j];
      if (v > bv) { bv = v; bi = j; }
    }
    red_v[tid] = bv; red_i[tid] = bi;
    __syncthreads();
    for (int s = 128; s > 0; s >>= 1) {
      if (tid < s && red_v[tid + s] > red_v[tid]) {
        red_v[tid] = red_v[tid + s]; red_i[tid] = red_i[tid + s];
      }
      __syncthreads();
    }
    if (tid == 0) {
      int w = red_i[0];
      top_s[b * K_RET + sel] = red_v[0];
      top_i[b * K_RET + sel] = (w >= 0) ? si[w] : -1;
      if (w >= 0) sc[w] = -INFINITY;
    }
    __syncthreads();
  }
}

// -------------------------------------------------------------------------
// Kernel 4: gather V_top, cross-attention softmax, weighted sum.
// One block (128 threads = 4 waves) per batch row.
// -------------------------------------------------------------------------
__global__ __launch_bounds__(128)
void attn_kernel(const float* __restrict__ qc, const float* __restrict__ em_V,
                 const float* __restrict__ top_s, const int* __restrict__ top_i,
                 float* __restrict__ out) {
  __shared__ __align__(16) float qcl[D_EMB];
  __shared__ __align__(16) float vtop[K_RET][D_EMB];
  __shared__ float lg[K_RET];
  __shared__ float wt[K_RET];
  const int b    = blockIdx.x;
  const int tid  = threadIdx.x;
  const int wave = tid >> 5;
  const int lane = tid & 31;

  qcl[tid] = qc[b * D_EMB + tid];
  for (int k = 0; k < K_RET; ++k) {
    int idx = top_i[b * K_RET + k];
    float v = 0.f;
    if (idx >= 0) v = em_V[((size_t)b * M_SLOTS + idx) * D_EMB + tid];
    vtop[k][tid] = v;
  }
  __syncthreads();

  // logits: wave w handles k = w, w+4, ...
  for (int j = 0; j < K_RET / 4; ++j) {
    const int k = wave + 4 * j;
    const float4 v4 = *reinterpret_cast<const float4*>(&vtop[k][lane * 4]);
    const float4 q4 = *reinterpret_cast<const float4*>(&qcl[lane * 4]);
    float p = q4.x * v4.x + q4.y * v4.y + q4.z * v4.z + q4.w * v4.w;
#pragma unroll
    for (int msk = 16; msk >= 1; msk >>= 1) p += __shfl_xor(p, msk, 32);
    if (lane == 0) {
      float ts = top_s[b * K_RET + k];
      lg[k] = (ts == -INFINITY) ? -INFINITY
                                : (p * 0.08838834764831845f + ts); // 128^-0.5
    }
  }
  __syncthreads();

  if (tid < K_RET) {
    float v = lg[tid];
    float m = v;
#pragma unroll
    for (int msk = 16; msk >= 1; msk >>= 1) {
      float o = __shfl_xor(m, msk, 32); m = (o > m) ? o : m;
    }
    float e = (m == -INFINITY || v == -INFINITY) ? 0.f : expf(v - m);
    float s = e;
#pragma unroll
    for (int msk = 16; msk >= 1; msk >>= 1) s += __shfl_xor(s, msk, 32);
    wt[tid] = (s > 0.f) ? (e / s) : 0.f;   // nan_to_num(softmax) semantics
  }
  __syncthreads();

  float acc = 0.f;
  for (int k = 0; k < K_RET; ++k) acc += wt[k] * vtop[k][tid];
  out[b * D_EMB + tid] = acc;
}

// -------------------------------------------------------------------------
// Kernel 5: LayerNorm -> FFN(gelu exact) -> residual -> @Wo + bo.
// One block, 16 waves, all GEMMs via V_WMMA_F32_16X16X4_F32. The 512-wide
// hidden is processed in four 128-column chunks staged through LDS.
// -------------------------------------------------------------------------
__global__ __launch_bounds__(512)
void ffn_out_kernel(const float* __restrict__ aout,
                    const float* __restrict__ ln_g, const float* __restrict__ ln_b,
                    const float* __restrict__ w1, const float* __restrict__ b1,
                    const float* __restrict__ w2, const float* __restrict__ b2,
                    const float* __restrict__ Wo, const float* __restrict__ bo,
                    float* __restrict__ out) {
  __shared__ float ob[BS][D_EMB];
  __shared__ float h[BS][D_EMB];     // LN result; later reused as y
  __shared__ float t1[BS][D_EMB];    // gelu(h @ w1_chunk)
  __shared__ float mu_s[BS], rs_s[BS];
  const int tid  = threadIdx.x;
  const int wave = tid >> 5;
  const int lane = tid & 31;
  const int half = lane >> 4;
  const int l16  = lane & 15;
  const int mt   = wave & 1;
  const int nt   = wave >> 1;
  const int m    = mt * 16 + l16;
  const int n    = nt * 16 + l16;

  for (int i = tid; i < BS * D_EMB; i += 512) ob[i / D_EMB][i % D_EMB] = aout[i];
  __syncthreads();
  if (tid < BS) {
    float mu = 0.f;
    for (int d = 0; d < D_EMB; ++d) mu += ob[tid][d];
    mu *= (1.f / D_EMB);
    float var = 0.f;
    for (int d = 0; d < D_EMB; ++d) { float z = ob[tid][d] - mu; var += z * z; }
    var *= (1.f / D_EMB);
    mu_s[tid] = mu; rs_s[tid] = rsqrtf(var + 1e-5f);
  }
  __syncthreads();
  for (int i = tid; i < BS * D_EMB; i += 512) {
    int r = i / D_EMB, c = i % D_EMB;
    h[r][c] = (ob[r][c] - mu_s[r]) * rs_s[r] * ln_g[c] + ln_b[c];
  }
  __syncthreads();

  v8f acc2 = {};
  for (int c = 0; c < 4; ++c) {
    // t1_chunk = gelu(h @ w1[:, c*128 .. c*128+128) + b1)
    v8f acc1 = {};
    for (int k = 0; k < D_EMB; k += 4) {
      v2f a, bb;
      a.x  = h[m][k + 2 * half + 0];
      a.y  = h[m][k + 2 * half + 1];
      bb.x = w1[(size_t)(k + 2 * half + 0) * 512 + c * 128 + n];
      bb.y = w1[(size_t)(k + 2 * half + 1) * 512 + c * 128 + n];
      acc1 = wmma_step(a, bb, acc1);
    }
    const float bias1 = b1[c * 128 + n];
#pragma unroll
    for (int r = 0; r < 8; ++r) {
      float v = acc1[r] + bias1;
      v = 0.5f * v * (1.f + erff(v * 0.70710678118654752f));  // exact GELU
      t1[mt * 16 + half * 8 + r][n] = v;
    }
    __syncthreads();
    // acc2 += t1_chunk @ w2[c*128 .. c*128+128, :]
    for (int k = 0; k < D_EMB; k += 4) {
      v2f a, bb;
      a.x  = t1[m][k + 2 * half + 0];
      a.y  = t1[m][k + 2 * half + 1];
      bb.x = w2[(size_t)(c * 128 + k + 2 * half + 0) * D_EMB + n];
      bb.y = w2[(size_t)(c * 128 + k + 2 * half + 1) * D_EMB + n];
      acc2 = wmma_step(a, bb, acc2);
    }
    __syncthreads();
  }

  // y = ffn2 + bias + residual  (reuse h)
  {
    const float bias2 = b2[n];
#pragma unroll
    for (int r = 0; r < 8; ++r) {
      int row = mt * 16 + half * 8 + r;
      h[row][n] = acc2[r] + bias2 + ob[row][n];
    }
  }
  __syncthreads();

  // out = y @ Wo[128,1024] + bo : 2x64 tiles, 8 per wave
  for (int tt = 0; tt < 8; ++tt) {
    const int t   = wave + 16 * tt;   // 0..127
    const int mt2 = t >> 6;
    const int nt2 = t & 63;
    const int mm  = mt2 * 16 + l16;
    const int nn  = nt2 * 16 + l16;
    v8f acc = {};
    for (int k = 0; k < D_EMB; k += 4) {
      v2f a, bb;
      a.x  = h[mm][k + 2 * half + 0];
      a.y  = h[mm][k + 2 * half + 1];
      bb.x = Wo[(size_t)(k + 2 * half + 0) * D_IN + nn];
      bb.y = Wo[(size_t)(k + 2 * half + 1) * D_IN + nn];
      acc = wmma_step(a, bb, acc);
    }
    const float bias = bo[nn];
#pragma unroll
    for (int r = 0; r < 8; ++r)
      out[(size_t)(mt2 * 16 + half * 8 + r) * D_IN + nn] = acc[r] + bias;
  }
}

// -------------------------------------------------------------------------
extern "C" void kernel_launch(void* const* d_in, const int* in_sizes, int n_in,
                              void* d_out, int out_size, void* d_ws, size_t ws_size,
                              hipStream_t stream) {
  const float* x     = (const float*)d_in[0];
  const float* y_wm  = (const float*)d_in[1];
  const float* em_K  = (const float*)d_in[2];
  const float* em_V  = (const float*)d_in[3];
  const float* em_S  = (const float*)d_in[4];
  const float* Wq_w  = (const float*)d_in[5];
  const float* Wq_b  = (const float*)d_in[6];
  const float* Wc_w  = (const float*)d_in[7];
  const float* Wc_b  = (const float*)d_in[8];
  const float* Wo_w  = (const float*)d_in[9];
  const float* Wo_b  = (const float*)d_in[10];
  const float* ln_g  = (const float*)d_in[11];
  const float* ln_b  = (const float*)d_in[12];
  const float* f1_w  = (const float*)d_in[13];
  const float* f1_b  = (const float*)d_in[14];
  const float* f2_w  = (const float*)d_in[15];
  const float* f2_b  = (const float*)d_in[16];

  float* ws     = (float*)d_ws;
  float* q      = ws;                         // 32*128
  float* qc     = q + BS * D_EMB;             // 32*128
  float* cand_s = qc + BS * D_EMB;            // 32*16*32
  int*   cand_i = (int*)(cand_s + BS * NBLK_M * K_RET);
  float* top_s  = (float*)(cand_i + BS * NBLK_M * K_RET);  // 32*32
  int*   top_i  = (int*)(top_s + BS * K_RET);              // 32*32
  float* aout   = (float*)(top_i + BS * K_RET);            // 32*128

  proj_kernel<<<1, 512, 0, stream>>>(x, y_wm, Wq_w, Wq_b, Wc_w, Wc_b, q, qc);
  scores_topk_kernel<<<dim3(NBLK_M, BS), 256, 0, stream>>>(q, em_K, em_S, cand_s, cand_i);
  merge_topk_kernel<<<BS, 256, 0, stream>>>(cand_s, cand_i, top_s, top_i);
  attn_kernel<<<BS, 128, 0, stream>>>(qc, em_V, top_s, top_i, aout);
  ffn_out_kernel<<<1, 512, 0, stream>>>(aout, ln_g, ln_b, f1_w, f1_b, f2_w, f2_b,
                                        Wo_w, Wo_b, (float*)d_out);
}